// V_Social_Aggregator_60962765800156
// MI455X (gfx1250) — compile-verified
//
#include <hip/hip_runtime.h>

#define LANES 32
#define WPB   8                 // waves per block (wave32 on gfx1250)
#define BLOCK (LANES * WPB)
#define EMBD  128               // embedding dim
#define CHUNK 4                 // rows per async double-buffer stage

#if defined(__AMDGCN__) && __has_builtin(__builtin_amdgcn_global_load_async_to_lds_b128) && \
    __has_builtin(__builtin_amdgcn_s_wait_asynccnt)
#define USE_ASYNC 1
#else
#define USE_ASYNC 0
#endif

// Pointer/vector types matching the async-LDS builtin signature:
// param0: v4i in addrspace(1) (global), param1: v4i in addrspace(3) (LDS).
typedef int v4i __attribute__((vector_size(16)));
typedef __attribute__((address_space(1))) v4i gv4i;
typedef __attribute__((address_space(3))) v4i lv4i;

// Wave-uniform lower_bound over sorted int32 array. The loaded value is pinned
// to an SGPR via readfirstlane so the whole loop stays on the SALU.
__device__ __forceinline__ int lower_bound_i32(const int* __restrict__ a, int n, int key) {
    int lo = 0, hi = n;
    while (lo < hi) {
        int mid = (lo + hi) >> 1;
        int v = __builtin_amdgcn_readfirstlane(a[mid]);
        if (v < key) lo = mid + 1; else hi = mid;
    }
    return lo;
}

__global__ __launch_bounds__(BLOCK) void seg_mean_kernel(
    const float* __restrict__ weight,     // [NUM_EMB, 128]
    const int*   __restrict__ neigh_idx,  // [E]
    const int*   __restrict__ seg_ids,    // [E], sorted
    float*       __restrict__ out,        // [N, 128]
    int n_edges, int n_nodes)
{
    const int lane = threadIdx.x & (LANES - 1);
    // threadIdx.x>>5 is wave-uniform on wave32; force it into an SGPR so the
    // binary search / index loads / loop bounds scalarize (SALU + s_load).
    const int w    = __builtin_amdgcn_readfirstlane(threadIdx.x >> 5);
    const int node = blockIdx.x * WPB + w;
    if (node >= n_nodes) return;

    const int lo  = lower_bound_i32(seg_ids, n_edges, node);
    const int hi  = lower_bound_i32(seg_ids, n_edges, node + 1);
    const int cnt = hi - lo;

    float ax = 0.f, ay = 0.f, az = 0.f, aw = 0.f;

#if USE_ASYNC
    // Double-buffered async gather: global -> LDS, ASYNCcnt-tracked, no VGPR
    // destinations in flight. 8 waves * 2 bufs * 4 rows * 512B = 32 KB LDS.
    __shared__ float4 stage[WPB][2][CHUNK][LANES];

    const int nch = (cnt + CHUNK - 1) / CHUNK;

    auto issue_chunk = [&](int c) {
        const int buf  = c & 1;
        const int base = lo + c * CHUNK;
#pragma unroll
        for (int r = 0; r < CHUNK; ++r) {
            const int e   = base + r;
            // Pad tail chunks with a safe duplicate row so every chunk issues
            // exactly CHUNK async ops (keeps the s_wait immediates sound).
            const int row = __builtin_amdgcn_readfirstlane((e < hi) ? neigh_idx[e]
                                                                    : neigh_idx[lo]);
            float* gsrc = const_cast<float*>(weight) + (size_t)row * EMBD + lane * 4;
            __builtin_amdgcn_global_load_async_to_lds_b128(
                (gv4i*)gsrc,
                (lv4i*)&stage[w][buf][r][lane],
                0, 0);
        }
    };

    if (nch > 0) {
        issue_chunk(0);
        for (int c = 0; c < nch; ++c) {
            if (c + 1 < nch) {
                issue_chunk(c + 1);
                // In-order completion: <= CHUNK outstanding means chunk c landed.
                __builtin_amdgcn_s_wait_asynccnt(CHUNK);
            } else {
                __builtin_amdgcn_s_wait_asynccnt(0);
            }
            const int buf  = c & 1;
            const int base = lo + c * CHUNK;
#pragma unroll
            for (int r = 0; r < CHUNK; ++r) {
                if (base + r < hi) {
                    float4 v = stage[w][buf][r][lane];
                    ax += v.x; ay += v.y; az += v.z; aw += v.w;
                }
            }
        }
    }
#else
    // Register-path fallback: 4 rows in flight per wave via manual unroll.
    float bx = 0.f, by = 0.f, bz = 0.f, bw = 0.f;
    float cx = 0.f, cy = 0.f, cz = 0.f, cw = 0.f;
    float dx = 0.f, dy = 0.f, dz = 0.f, dw = 0.f;
    int e = lo;
    for (; e + 4 <= hi; e += 4) {
        const int r0 = __builtin_amdgcn_readfirstlane(neigh_idx[e + 0]);
        const int r1 = __builtin_amdgcn_readfirstlane(neigh_idx[e + 1]);
        const int r2 = __builtin_amdgcn_readfirstlane(neigh_idx[e + 2]);
        const int r3 = __builtin_amdgcn_readfirstlane(neigh_idx[e + 3]);
        float4 v0 = *(const float4*)(weight + (size_t)r0 * EMBD + lane * 4);
        float4 v1 = *(const float4*)(weight + (size_t)r1 * EMBD + lane * 4);
        float4 v2 = *(const float4*)(weight + (size_t)r2 * EMBD + lane * 4);
        float4 v3 = *(const float4*)(weight + (size_t)r3 * EMBD + lane * 4);
        ax += v0.x; ay += v0.y; az += v0.z; aw += v0.w;
        bx += v1.x; by += v1.y; bz += v1.z; bw += v1.w;
        cx += v2.x; cy += v2.y; cz += v2.z; cw += v2.w;
        dx += v3.x; dy += v3.y; dz += v3.z; dw += v3.w;
    }
    for (; e < hi; ++e) {
        const int r0 = __builtin_amdgcn_readfirstlane(neigh_idx[e]);
        float4 v0 = *(const float4*)(weight + (size_t)r0 * EMBD + lane * 4);
        ax += v0.x; ay += v0.y; az += v0.z; aw += v0.w;
    }
    ax += bx + cx + dx; ay += by + cy + dy;
    az += bz + cz + dz; aw += bw + cw + dw;
#endif

    // Mean with max(count,1) semantics (zero-count segments output zeros).
    const float cdiv = (cnt > 0) ? (float)cnt : 1.0f;
    float4 res;
    res.x = ax / cdiv; res.y = ay / cdiv; res.z = az / cdiv; res.w = aw / cdiv;
    *(float4*)(out + (size_t)node * EMBD + lane * 4) = res;
}

extern "C" void kernel_launch(void* const* d_in, const int* in_sizes, int n_in,
                              void* d_out, int out_size, void* d_ws, size_t ws_size,
                              hipStream_t stream) {
    const float* weight = (const float*)d_in[0];
    const int*   neigh  = (const int*)d_in[1];
    const int*   segs   = (const int*)d_in[2];
    float*       out    = (float*)d_out;

    const int n_edges = in_sizes[1];
    const int n_nodes = out_size / EMBD;   // 10000 * 128 floats out

    const int grid = (n_nodes + WPB - 1) / WPB;
    hipLaunchKernelGGL(seg_mean_kernel, dim3(grid), dim3(BLOCK), 0, stream,
                       weight, neigh, segs, out, n_edges, n_nodes);
}